// GIN_Model_79680233276330
// MI455X (gfx1250) — compile-verified
//
#include <hip/hip_runtime.h>
#include <hip/hip_bf16.h>
#include <math.h>

typedef __attribute__((ext_vector_type(2))) float v2f;
typedef __attribute__((ext_vector_type(8))) float v8f;

#define DHID 128
#define N_NODES 100000
#define N_EDGES 1600000
#define N_GRAPHS 256

// -------------------------------------------------------------------------
// Generic 128-wide WMMA fp32 GEMM:
//   out[m][n] = act( (A0[m][k] + alpha*A1[m][k]) * W[n][k] + bias[n] )
// alpha = 1 + eps (fused GIN combine), act = relu if RELU.
// Block: 256 threads (8 waves). Wave w computes rows [blk*128 + w*16, +16),
// all 128 columns, as 8 x V_WMMA_F32_16X16X4_F32 accumulators.
//
// LDS holds W^T in a "k-paired" layout so each B fragment {B[kk][n],
// B[kk+1][n]} is one aligned ds_load_b64 straight into an even VGPR pair:
//   wtp[(k>>1)*256 + n*2 + (k&1)] = W[n][k]        (64 KB)
// A rows are loaded with a clamped row index (no EXEC masking in the loop);
// out-of-range rows only pollute their own accumulator rows, never stored.
// -------------------------------------------------------------------------
template <bool HAS_A1, bool RELU>
__global__ __launch_bounds__(256, 1) void gemm_nt_wmma(
    float* __restrict__ out,
    const float* __restrict__ A0,
    const float* __restrict__ A1,       // second A operand (GIN self term)
    const float* __restrict__ epsPtr,   // alpha = 1 + *epsPtr
    const float* __restrict__ W,        // [128][128] row-major (out-dim major)
    const float* __restrict__ bias,     // [128]
    int M) {
  __shared__ float wtp[DHID * DHID];  // 64 KB, k-paired W^T
  for (int i = threadIdx.x; i < DHID * DHID; i += blockDim.x) {
    int n = i >> 7, k = i & 127;  // coalesced read of W[n][k]
    wtp[((k >> 1) << 8) + (n << 1) + (k & 1)] = W[i];
  }
  __syncthreads();

  const int lane = threadIdx.x & 31;
  const int wave = threadIdx.x >> 5;
  const int row0 = blockIdx.x * 128 + wave * 16;
  const int koff = (lane >> 4) << 1;           // A/B frag K split across halves
  const int ncol = lane & 15;
  int mrow = row0 + ncol;
  if (mrow > M - 1) mrow = M - 1;              // clamp: always-valid load row
  float alpha = 1.0f;
  if (HAS_A1) alpha = 1.0f + epsPtr[0];

  v8f acc[8] = {};  // 8 n-tiles of 16, fp32 C/D = 8 VGPRs each

  const float* aptr  = A0 + (size_t)mrow * DHID + koff;
  const float* a1ptr = HAS_A1 ? (A1 + (size_t)mrow * DHID + koff) : nullptr;
  // per-lane base into paired-B LDS: (koff>>1)*256 + ncol*2 (elements)
  const float* bbase = wtp + ((koff >> 1) << 8);

#pragma unroll 2
  for (int k = 0; k < DHID; k += 4) {
    v2f a = *(const v2f*)(aptr + k);           // A[mrow][kk..kk+1]
    if (HAS_A1) {
      v2f a1 = *(const v2f*)(a1ptr + k);
      a.x = fmaf(alpha, a1.x, a.x);
      a.y = fmaf(alpha, a1.y, a.y);
    }
    const float* brow = bbase + ((k >> 1) << 8);
#pragma unroll
    for (int nt = 0; nt < 8; ++nt) {
      v2f b = *(const v2f*)(brow + ((nt * 16 + ncol) << 1));  // ds_load_b64
      // D = A(16x4) x B(4x16) + C ; fp32 in, fp32 out
      acc[nt] = __builtin_amdgcn_wmma_f32_16x16x4_f32(
          /*neg_a=*/false, a, /*neg_b=*/false, b,
          /*c_mod=*/(short)0, acc[nt], /*reuse_a=*/false, /*reuse_b=*/false);
    }
  }

  // C/D layout: VGPR i -> M = i (lanes 0-15) or M = 8+i (lanes 16-31); N = ncol
  const int hi = (lane >> 4) * 8;
  if (row0 + 16 <= M) {  // fast path: whole tile in range, unguarded stores
#pragma unroll
    for (int nt = 0; nt < 8; ++nt) {
      const int n = nt * 16 + ncol;
      const float bn = bias[n];
#pragma unroll
      for (int i = 0; i < 8; ++i) {
        float v = acc[nt][i] + bn;
        if (RELU) v = fmaxf(v, 0.0f);
        out[(size_t)(row0 + hi + i) * DHID + n] = v;
      }
    }
  } else {
#pragma unroll
    for (int nt = 0; nt < 8; ++nt) {
      const int n = nt * 16 + ncol;
      const float bn = bias[n];
#pragma unroll
      for (int i = 0; i < 8; ++i) {
        int r = row0 + hi + i;
        if (r < M) {
          float v = acc[nt][i] + bn;
          if (RELU) v = fmaxf(v, 0.0f);
          out[(size_t)r * DHID + n] = v;
        }
      }
    }
  }
}

// -------------------------------------------------------------------------
// agg[dst[e]][d] += mask[e] * h[src[e]][d]   (hardware fp32 atomics, L2-resident)
// 256 threads = 2 edges x 128 dims, coalesced row read + coalesced atomics.
// -------------------------------------------------------------------------
__global__ __launch_bounds__(256) void scatter_edges(
    float* __restrict__ agg, const float* __restrict__ h,
    const int* __restrict__ dst, const int* __restrict__ src,
    const float* __restrict__ mask, int nEdges) {
  int e = blockIdx.x * 2 + (threadIdx.x >> 7);
  int d = threadIdx.x & 127;
  if (e < nEdges) {
    float m = mask[e];
    int s = src[e];
    int t = dst[e];
    float v = h[(size_t)s * DHID + d] * m;
    unsafeAtomicAdd(&agg[(size_t)t * DHID + d], v);
  }
}

// -------------------------------------------------------------------------
// pooled_sum[batch[n]][d] += sum_{e=0..4} embeds[e][n][d]
// One block per node: local 5-way sum, single atomic per element.
// -------------------------------------------------------------------------
__global__ __launch_bounds__(128) void pool_kernel(
    float* __restrict__ ps, const float* __restrict__ embeds,
    const int* __restrict__ batch, int nNodes) {
  int node = blockIdx.x;
  int d = threadIdx.x;
  int g = batch[node];
  float s = 0.0f;
#pragma unroll
  for (int e = 0; e < 5; ++e)
    s += embeds[((size_t)e * nNodes + node) * DHID + d];
  unsafeAtomicAdd(&ps[(size_t)g * DHID + d], s);
}

__global__ __launch_bounds__(256) void fill_zero(float* __restrict__ p, size_t n) {
  size_t i = (size_t)blockIdx.x * blockDim.x + threadIdx.x;
  if (i < n) p[i] = 0.0f;
}

// -------------------------------------------------------------------------
// Head: lin1_dropped = lin1 (dropout identity); lin2 = lin1 @ W2^T + b2;
// softmax. One thread per graph row (256 rows total).
// -------------------------------------------------------------------------
__global__ __launch_bounds__(128) void head_kernel(
    const float* __restrict__ lin1, float* __restrict__ lin1_drop,
    const float* __restrict__ W2, const float* __restrict__ b2,
    float* __restrict__ lin2_out, float* __restrict__ lin2_soft) {
  int r = blockIdx.x * blockDim.x + threadIdx.x;
  if (r >= N_GRAPHS) return;
  float xr[DHID];
#pragma unroll 8
  for (int k = 0; k < DHID; ++k) {
    float v = lin1[(size_t)r * DHID + k];
    xr[k] = v;
    lin1_drop[(size_t)r * DHID + k] = v;
  }
  float o[8];
  float mx = -INFINITY;
#pragma unroll
  for (int j = 0; j < 8; ++j) {
    float s = b2[j];
    for (int k = 0; k < DHID; ++k) s = fmaf(xr[k], W2[j * DHID + k], s);
    o[j] = s;
    lin2_out[r * 8 + j] = s;
    mx = fmaxf(mx, s);
  }
  float den = 0.0f;
#pragma unroll
  for (int j = 0; j < 8; ++j) {
    o[j] = __expf(o[j] - mx);
    den += o[j];
  }
  float inv = 1.0f / den;
#pragma unroll
  for (int j = 0; j < 8; ++j) lin2_soft[r * 8 + j] = o[j] * inv;
}

// -------------------------------------------------------------------------
extern "C" void kernel_launch(void* const* d_in, const int* in_sizes, int n_in,
                              void* d_out, int out_size, void* d_ws, size_t ws_size,
                              hipStream_t stream) {
  const float* x       = (const float*)d_in[0];
  const int*   eidx    = (const int*)d_in[1];
  const float* emask   = (const float*)d_in[2];
  const int*   batch   = (const int*)d_in[3];
  const float* eps     = (const float*)d_in[4];
  const float* W_first = (const float*)d_in[5];
  const float* b_first = (const float*)d_in[6];
  const float* gin_W   = (const float*)d_in[7];  // [4][2][128][128]
  const float* gin_b   = (const float*)d_in[8];  // [4][2][128]
  const float* W1      = (const float*)d_in[9];
  const float* b1      = (const float*)d_in[10];
  const float* W2      = (const float*)d_in[11];
  const float* b2      = (const float*)d_in[12];

  // d_out layout (reference return order, flat):
  float* out        = (float*)d_out;
  float* embeds     = out;                                   // [5][N][128]
  float* pooled_sum = out + (size_t)5 * N_NODES * DHID;      // [256][128]
  float* lin1_out   = pooled_sum + N_GRAPHS * DHID;          // [256][128]
  float* lin1_drop  = lin1_out + N_GRAPHS * DHID;            // [256][128]
  float* lin2_out   = lin1_drop + N_GRAPHS * DHID;           // [256][8]
  float* lin2_soft  = lin2_out + N_GRAPHS * 8;               // [256][8]

  // scratch: agg + tmp (51.2 MB each)
  float* agg = (float*)d_ws;
  float* tmp = agg + (size_t)N_NODES * DHID;

  const int* dst = eidx;            // edge_index[0]
  const int* src = eidx + N_EDGES;  // edge_index[1]

  const dim3 gemmGrid((N_NODES + 127) / 128);  // 782 blocks x 256 threads

  // 1) first linear embedding -> embeds[0]
  gemm_nt_wmma<false, false><<<gemmGrid, 256, 0, stream>>>(
      embeds, x, nullptr, nullptr, W_first, b_first, N_NODES);

  // 2) GIN layers
  for (int l = 0; l < 4; ++l) {
    const float* h = embeds + (size_t)l * N_NODES * DHID;
    float* hnext   = embeds + (size_t)(l + 1) * N_NODES * DHID;
    const size_t aggN = (size_t)N_NODES * DHID;

    fill_zero<<<(unsigned)((aggN + 255) / 256), 256, 0, stream>>>(agg, aggN);
    scatter_edges<<<(N_EDGES + 1) / 2, 256, 0, stream>>>(agg, h, dst, src, emask, N_EDGES);

    // SLP layer 0: relu((agg + (1+eps_l)*h) @ W^T + b), combine fused into A load
    gemm_nt_wmma<true, true><<<gemmGrid, 256, 0, stream>>>(
        tmp, agg, h, eps + l,
        gin_W + ((size_t)l * 2 + 0) * DHID * DHID, gin_b + (l * 2 + 0) * DHID, N_NODES);
    // SLP layer 1
    gemm_nt_wmma<false, true><<<gemmGrid, 256, 0, stream>>>(
        hnext, tmp, nullptr, nullptr,
        gin_W + ((size_t)l * 2 + 1) * DHID * DHID, gin_b + (l * 2 + 1) * DHID, N_NODES);
  }

  // 3) joint embeddings: sum of global add-pools of all 5 embeddings
  fill_zero<<<(N_GRAPHS * DHID + 255) / 256, 256, 0, stream>>>(pooled_sum, N_GRAPHS * DHID);
  pool_kernel<<<N_NODES, 128, 0, stream>>>(pooled_sum, embeds, batch, N_NODES);

  // 4) lin1 = relu(pooled_sum @ W1^T + b1)  (M=256 -> 2 blocks)
  gemm_nt_wmma<false, true><<<(N_GRAPHS + 127) / 128, 256, 0, stream>>>(
      lin1_out, pooled_sum, nullptr, nullptr, W1, b1, N_GRAPHS);

  // 5) dropout-identity copy + lin2 + softmax
  head_kernel<<<2, 128, 0, stream>>>(lin1_out, lin1_drop, W2, b2, lin2_out, lin2_soft);
}